// Sampler_6305011990586
// MI455X (gfx1250) — compile-verified
//
#include <hip/hip_runtime.h>
#include <hip/hip_bf16.h>

// ---------------- problem constants ----------------
#define NWAY   64
#define KSHOT  16
#define QSHOT  512
#define TOPK   20
#define DIM    768
#define NSUP   (NWAY * KSHOT)       // 1024 support rows
#define NQ     (NWAY * QSHOT)       // 32768 query rows

// ---------------- tiling constants -----------------
#define CHUNK   2048                // columns per block (top-k chunk)
#define NCHUNK  (NQ / CHUNK)        // 16
#define QTILE   128                 // queries staged per iteration (8 waves * 16)
#define KTILE   64                  // k-depth staged per iteration
#define NKB     (DIM / KTILE)       // 12 k-blocks
#define ASTR    (DIM + 4)           // LDS stride for A tile (bank-conflict pad)
#define BSTR    (KTILE + 4)         // LDS stride for B tile [q][k] (pad: 68)
#define DSTR    (CHUNK + 16)        // LDS stride for distance tile

typedef float v2f __attribute__((ext_vector_type(2)));
typedef float v8f __attribute__((ext_vector_type(8)));
typedef int   v4i __attribute__((vector_size(4 * sizeof(int))));

// ---- CDNA5 async global->LDS path (guarded; falls back to sync copy) ----
#if defined(__has_builtin)
#if __has_builtin(__builtin_amdgcn_global_load_async_to_lds_b128) && \
    __has_builtin(__builtin_amdgcn_s_wait_asynccnt)
#define USE_ASYNC 1
#endif
#endif

#ifdef USE_ASYNC
typedef __attribute__((address_space(1))) v4i gv4i;   // global-AS 16B vector
typedef __attribute__((address_space(3))) v4i lv4i;   // LDS-AS 16B vector
#define WAIT_ASYNC(n) __builtin_amdgcn_s_wait_asynccnt(n)
#else
#define WAIT_ASYNC(n)
#endif

__device__ __forceinline__ void copy16_to_lds(const float* g, float* l) {
#ifdef USE_ASYNC
    __builtin_amdgcn_global_load_async_to_lds_b128(
        (gv4i*)g, (lv4i*)l, 0, 0);
#else
    *(float4*)l = *(const float4*)g;
#endif
}

// monotone float -> uint mapping: order-preserving for all finite floats
__device__ __forceinline__ unsigned int orderedBits(float f) {
    unsigned int u = __float_as_uint(f);
    return (u & 0x80000000u) ? ~u : (u | 0x80000000u);
}

// ---------------------------------------------------------------------------
// Kernel A: squared row norms. One wave per row, 8 rows per 256-thread block.
// ---------------------------------------------------------------------------
__global__ __launch_bounds__(256) void rownorm_kernel(const float* __restrict__ x,
                                                      float* __restrict__ out,
                                                      int nrows) {
    const int wave = threadIdx.x >> 5;
    const int lane = threadIdx.x & 31;
    const int row  = blockIdx.x * 8 + wave;
    if (row >= nrows) return;
    const float* p = x + (size_t)row * DIM;
    float s = 0.0f;
    for (int i = lane * 4; i < DIM; i += 32 * 4) {
        const float4 v = *(const float4*)(p + i);
        s += v.x * v.x + v.y * v.y + v.z * v.z + v.w * v.w;
    }
    for (int sft = 16; sft > 0; sft >>= 1) s += __shfl_xor(s, sft, 32);
    if (lane == 0) out[row] = s;
}

// ---------------------------------------------------------------------------
// Kernel B: fused fp32-WMMA distance GEMM + per-chunk partial top-20.
//   grid = (NSUP/16, NCHUNK), block = 256 (8 waves).
//   Each wave computes a 16x16 D-tile with V_WMMA_F32_16X16X4_F32.
//   B tiles are double-buffered and staged with async global->LDS DMA.
// ---------------------------------------------------------------------------
__global__ __launch_bounds__(256, 1) void distance_topk_kernel(
    const float* __restrict__ sup, const float* __restrict__ qry,
    const float* __restrict__ x2g, const float* __restrict__ y2g,
    unsigned long long* __restrict__ partKeys) {

    __shared__ float As[16 * ASTR];         //  49,408 B : support tile (full K)
    __shared__ float Bs[2][QTILE * BSTR];   //  69,632 B : query tiles [q][k]
    __shared__ float Ds[16 * DSTR];         // 132,096 B : distance tile
    __shared__ float x2s[16];

    const int rowBlock  = blockIdx.x;        // 0..63
    const int chunkId   = blockIdx.y;        // 0..NCHUNK-1
    const int chunkBase = chunkId * CHUNK;
    const int tid  = threadIdx.x;
    const int lane = tid & 31;
    const int wave = tid >> 5;               // 0..7

    // ---- stage A (16 support rows x 768), float4 coalesced ----
    for (int i = tid; i < 16 * (DIM / 4); i += 256) {
        const int r  = i / (DIM / 4);
        const int c4 = (i % (DIM / 4)) * 4;
        const float4 v = *(const float4*)(sup + (size_t)(rowBlock * 16 + r) * DIM + c4);
        As[r * ASTR + c4 + 0] = v.x;
        As[r * ASTR + c4 + 1] = v.y;
        As[r * ASTR + c4 + 2] = v.z;
        As[r * ASTR + c4 + 3] = v.w;
    }
    if (tid < 16) x2s[tid] = x2g[rowBlock * 16 + tid];
    __syncthreads();

    const int mrow = lane & 15;              // A row / also B column index
    const int hi   = lane >> 4;              // half-wave selector (K offset +2)

    // ---- GEMM over this chunk, QTILE queries at a time ----
    for (int qb = 0; qb < CHUNK; qb += QTILE) {
        // stage: 128 queries x 64 dims, contiguous 16B per thread-chunk
        auto stage = [&](int buf, int kb) {
            for (int it = 0; it < 8; ++it) {
                const int chunk = tid + it * 256;        // 0..2047
                const int q  = chunk >> 4;               // 0..127
                const int kc = (chunk & 15) * 4;         // 0..60
                copy16_to_lds(qry + (size_t)(chunkBase + qb + q) * DIM + kb + kc,
                              &Bs[buf][q * BSTR + kc]);
            }
        };

        v8f c = {};                          // 16x16 fp32 accumulator (8 VGPRs)
        stage(0, 0);                         // prime the pipeline
        for (int kbi = 0; kbi < NKB; ++kbi) {
            const int cur = kbi & 1;
            if (kbi + 1 < NKB) {
                stage(cur ^ 1, (kbi + 1) * KTILE);       // overlap with compute
                WAIT_ASYNC(8);               // retire current buffer only
            } else {
                WAIT_ASYNC(0);
            }
            __syncthreads();                 // all waves' DMA of cur visible

            const int kb   = kbi * KTILE;
            const int qOff = wave * 16 + mrow;
#pragma unroll
            for (int k4 = 0; k4 < KTILE; k4 += 4) {
                // A fragment: lanes 0-15 -> K = k4+0,+1 ; lanes 16-31 -> +2,+3
                const float2 av = *(const float2*)&As[mrow * ASTR + kb + k4 + hi * 2];
                v2f a; a.x = av.x; a.y = av.y;
                // B fragment (layout [q][k]): same K pairing per half-wave
                const float2 bv = *(const float2*)&Bs[cur][qOff * BSTR + k4 + hi * 2];
                v2f b; b.x = bv.x; b.y = bv.y;
                c = __builtin_amdgcn_wmma_f32_16x16x4_f32(
                        false, a, false, b, (short)0, c, false, false);
            }
            __syncthreads();                 // before buffer cur is re-filled
        }
        // ---- distances into LDS: D[m][col] = x2[m] + y2[col] - 2*dot ----
        const float y2v = y2g[chunkBase + qb + wave * 16 + mrow];
#pragma unroll
        for (int i = 0; i < 8; ++i) {
            const int m = i + hi * 8;
            Ds[m * DSTR + qb + wave * 16 + mrow] = x2s[m] + y2v - 2.0f * c[i];
        }
    }
    __syncthreads();

    // ---- partial top-20 per row (2 rows per wave), 20 argmin passes ----
    for (int r = 0; r < 2; ++r) {
        const int row = wave * 2 + r;
        for (int t = 0; t < TOPK; ++t) {
            unsigned long long best = ~0ull;
            for (int j = lane; j < CHUNK; j += 32) {
                const unsigned long long key =
                    ((unsigned long long)orderedBits(Ds[row * DSTR + j]) << 32) |
                    (unsigned int)(chunkBase + j);
                best = key < best ? key : best;
            }
            for (int s = 16; s > 0; s >>= 1) {
                const unsigned long long o = __shfl_xor(best, s, 32);
                best = o < best ? o : best;
            }
            if (lane == 0) {
                partKeys[((size_t)(rowBlock * 16 + row) * NCHUNK + chunkId) * TOPK + t] = best;
                const int col = (int)(best & 0xFFFFFFFFull) - chunkBase;
                Ds[row * DSTR + col] = __int_as_float(0x7F800000);  // +inf
            }
            __syncthreads();
        }
    }
}

// ---------------------------------------------------------------------------
// Kernel C: merge NCHUNK partial top-20 lists -> final 20 + per-row correct
//   count. One wave per row. Key order matches jax.lax.top_k tie-breaking.
// ---------------------------------------------------------------------------
__global__ __launch_bounds__(32) void merge_topk_kernel(
    const unsigned long long* __restrict__ partKeys,
    int* __restrict__ finalIdx, int* __restrict__ rowCorrect) {

    __shared__ unsigned long long keys[NCHUNK * TOPK];   // 320 candidates
    const int row  = blockIdx.x;
    const int lane = threadIdx.x;

    for (int i = lane; i < NCHUNK * TOPK; i += 32)
        keys[i] = partKeys[(size_t)row * NCHUNK * TOPK + i];
    __syncthreads();

    const int cls = row / KSHOT;
    int correct = 0;
    for (int t = 0; t < TOPK; ++t) {
        unsigned long long best = ~0ull;
        int bi = 0;
        for (int i = lane; i < NCHUNK * TOPK; i += 32) {
            const unsigned long long k = keys[i];
            if (k < best) { best = k; bi = i; }
        }
        for (int s = 16; s > 0; s >>= 1) {
            const unsigned long long ob = __shfl_xor(best, s, 32);
            const int obi = __shfl_xor(bi, s, 32);
            if (ob < best) { best = ob; bi = obi; }
        }
        if (lane == 0) {
            const int idx = (int)(best & 0xFFFFFFFFull);
            finalIdx[row * TOPK + t] = idx;
            if (idx / QSHOT == cls) correct++;
            keys[bi] = ~0ull;
        }
        __syncthreads();
    }
    if (lane == 0) rowCorrect[row] = correct;
}

// ---------------------------------------------------------------------------
// Kernel D: accuracy scalar (deterministic integer tree reduction).
// ---------------------------------------------------------------------------
__global__ __launch_bounds__(NSUP) void acc_kernel(const int* __restrict__ rowCorrect,
                                                   float* __restrict__ accOut) {
    __shared__ int wsum[32];
    const int tid = threadIdx.x;
    int v = rowCorrect[tid];
    for (int s = 16; s > 0; s >>= 1) v += __shfl_xor(v, s, 32);
    if ((tid & 31) == 0) wsum[tid >> 5] = v;
    __syncthreads();
    if (tid < 32) {
        int t = wsum[tid];
        for (int s = 16; s > 0; s >>= 1) t += __shfl_xor(t, s, 32);
        if (tid == 0) accOut[0] = (float)t / (float)(NWAY * KSHOT * TOPK);
    }
}

// ---------------------------------------------------------------------------
// Kernel E: gather query rows for the selected indices (float4 copies).
//   grid = NSUP*TOPK blocks, block = DIM/4 = 192 threads.
// ---------------------------------------------------------------------------
__global__ __launch_bounds__(DIM / 4) void gather_kernel(
    const float* __restrict__ qry, const int* __restrict__ finalIdx,
    float* __restrict__ out) {
    const int b   = blockIdx.x;              // row*TOPK + j
    const int idx = finalIdx[b];
    const float4* src = (const float4*)(qry + (size_t)idx * DIM);
    float4*       dst = (float4*)(out + (size_t)b * DIM);
    dst[threadIdx.x] = src[threadIdx.x];
}

// ---------------------------------------------------------------------------
extern "C" void kernel_launch(void* const* d_in, const int* in_sizes, int n_in,
                              void* d_out, int out_size, void* d_ws, size_t ws_size,
                              hipStream_t stream) {
    const float* sup = (const float*)d_in[0];   // [1024, 768]
    const float* qry = (const float*)d_in[1];   // [32768, 768]
    float* out = (float*)d_out;                 // [1024*20*768] sampled + [1] acc

    // workspace layout (~2.84 MB)
    float* x2 = (float*)d_ws;                                   // 1024 f
    float* y2 = x2 + NSUP;                                      // 32768 f
    unsigned long long* partKeys = (unsigned long long*)(y2 + NQ);  // 8B aligned
    int* finalIdx   = (int*)(partKeys + (size_t)NSUP * NCHUNK * TOPK);
    int* rowCorrect = finalIdx + NSUP * TOPK;

    rownorm_kernel<<<NSUP / 8, 256, 0, stream>>>(sup, x2, NSUP);
    rownorm_kernel<<<NQ / 8, 256, 0, stream>>>(qry, y2, NQ);

    dim3 grid(NSUP / 16, NCHUNK);
    distance_topk_kernel<<<grid, 256, 0, stream>>>(sup, qry, x2, y2, partKeys);

    merge_topk_kernel<<<NSUP, 32, 0, stream>>>(partKeys, finalIdx, rowCorrect);
    acc_kernel<<<1, NSUP, 0, stream>>>(rowCorrect, out + (size_t)NSUP * TOPK * DIM);
    gather_kernel<<<NSUP * TOPK, DIM / 4, 0, stream>>>(qry, finalIdx, out);
}